// GCN_85899346397
// MI455X (gfx1250) — compile-verified
//
#include <hip/hip_runtime.h>

// ---------------------------------------------------------------------------
// GCN (3x GCNConv + linear classifier) for MI455X / gfx1250, fp32 WMMA path.
// Dense transforms use V_WMMA_F32_16X16X4_F32 (one 16-row tile per wave32);
// sparse aggregation uses L2-resident f32 atomics (feature maps fit in 192MB L2).
// ---------------------------------------------------------------------------

typedef float v2f __attribute__((ext_vector_type(2)));
typedef float v8f __attribute__((ext_vector_type(8)));

__global__ void k_set1(float* __restrict__ p, int n) {
  int i = blockIdx.x * blockDim.x + threadIdx.x;
  if (i < n) p[i] = 1.0f;   // self-loop contributes 1 to every degree
}

__global__ void k_zero(float* __restrict__ p, int n) {
  int i = blockIdx.x * blockDim.x + threadIdx.x;
  if (i < n) p[i] = 0.0f;
}

// deg[col[e]] += 1  (E real edges; self-loops pre-seeded by k_set1)
__global__ void k_deg(const long long* __restrict__ col, float* __restrict__ deg, int E) {
  int e = blockIdx.x * blockDim.x + threadIdx.x;
  if (e < E) atomicAdd(deg + (int)col[e], 1.0f);
}

__global__ void k_rsqrt(float* __restrict__ d, int n) {
  int i = blockIdx.x * blockDim.x + threadIdx.x;
  if (i < n) d[i] = rsqrtf(d[i]);   // deg >= 1 always (self-loop)
}

// ---------------------------------------------------------------------------
// H[nrows x NOUT] = X[nrows x K] @ W[K x NOUT] (+ bias), fp32 WMMA 16x16x4.
// One wave per (16-row, 16-col) output tile. W staged in LDS.
// ISA layouts (CDNA5 ISA §7.12.2):
//   A 16x4 f32 : lane<16 holds M=lane,K={k,k+1}; lane>=16 holds M=lane-16,K={k+2,k+3}
//   B  4x16 f32: lane<16 holds N=lane rows K={k,k+1}; lane>=16 rows K={k+2,k+3}
//   C/D 16x16  : VGPR j = rows M=j (lanes 0-15) and M=j+8 (lanes 16-31)
// nrows must be a multiple of 16 (N = 100000 = 6250*16).
// ---------------------------------------------------------------------------
template <int K, int NOUT>
__global__ __launch_bounds__(128) void k_gemm_wmma(
    const float* __restrict__ X, const float* __restrict__ W,
    const float* __restrict__ bias, float* __restrict__ H, int nrows) {
  __shared__ float sW[K * NOUT];
  for (int i = threadIdx.x; i < K * NOUT; i += blockDim.x) sW[i] = W[i];
  __syncthreads();

  const int lane = threadIdx.x & 31;
  const int wid  = blockIdx.x * (blockDim.x >> 5) + (threadIdx.x >> 5);
  constexpr int NT = NOUT / 16;
  const int mt   = wid / NT;
  const int nt   = wid - mt * NT;
  const int row0 = mt * 16;
  if (row0 >= nrows) return;
  const int n0 = nt * 16;
  const int m  = lane & 15;          // row (A) / col (B,D) within tile
  const int kh = (lane >> 4) << 1;   // 0 for lanes 0-15, 2 for lanes 16-31

  const float* __restrict__ xrow = X + (size_t)(row0 + m) * K;
  v8f c = {};
  for (int k = 0; k < K; k += 4) {
    v2f a, b;
    a[0] = xrow[k + kh];
    a[1] = xrow[k + kh + 1];
    b[0] = sW[(k + kh) * NOUT + n0 + m];
    b[1] = sW[(k + kh + 1) * NOUT + n0 + m];
    c = __builtin_amdgcn_wmma_f32_16x16x4_f32(
        /*neg_a=*/false, a, /*neg_b=*/false, b,
        /*c_mod=*/(short)0, c, /*reuse_a=*/false, /*reuse_b=*/false);
  }

  const float bv = bias ? bias[n0 + m] : 0.0f;
  const int rb = row0 + ((lane >> 4) << 3);   // rows j / j+8 split across lane halves
#pragma unroll
  for (int j = 0; j < 8; ++j)
    H[(size_t)(rb + j) * NOUT + n0 + m] = c[j] + bv;
}

// agg[col[e], f] += h[row[e], f] * dis[row[e]] * dis[col[e]]
// One thread per (edge, feature): coalesced 64/128B feature vectors, L2 atomics.
template <int F>
__global__ void k_scatter(const float* __restrict__ h,
                          const long long* __restrict__ row,
                          const long long* __restrict__ col,
                          const float* __restrict__ dis,
                          float* __restrict__ agg, long long E) {
  long long t = (long long)blockIdx.x * blockDim.x + threadIdx.x;
  long long e = t / F;
  if (e >= E) return;
  int f = (int)(t - e * F);
  int r = (int)row[e];
  int c = (int)col[e];
  float w = dis[r] * dis[c];
  atomicAdd(&agg[(size_t)c * F + f], h[(size_t)r * F + f] * w);
}

// out = relu(agg + h * dis^2 + b)   (self-loop term fused: norm_ii = dis_i^2)
template <int F>
__global__ void k_biasrelu(const float* __restrict__ agg, const float* __restrict__ h,
                           const float* __restrict__ dis, const float* __restrict__ b,
                           float* __restrict__ out, int n) {
  int t = blockIdx.x * blockDim.x + threadIdx.x;
  int i = t / F;
  if (i >= n) return;
  int f = t - i * F;
  float d = dis[i];
  float v = agg[t] + h[t] * d * d + b[f];
  out[t] = fmaxf(v, 0.0f);
}

static inline int cdiv(long long a, long long b) { return (int)((a + b - 1) / b); }

extern "C" void kernel_launch(void* const* d_in, const int* in_sizes, int n_in,
                              void* d_out, int out_size, void* d_ws, size_t ws_size,
                              hipStream_t stream) {
  const float*     x  = (const float*)d_in[0];
  const long long* ei = (const long long*)d_in[1];     // int64 [2, E]
  const float* W1 = (const float*)d_in[2];
  const float* b1 = (const float*)d_in[3];
  const float* W2 = (const float*)d_in[4];
  const float* b2 = (const float*)d_in[5];
  const float* W3 = (const float*)d_in[6];
  const float* b3 = (const float*)d_in[7];
  const float* Wc = (const float*)d_in[8];
  const float* bc = (const float*)d_in[9];

  const int n = in_sizes[0] / 512;      // 100000 (multiple of 16)
  const int E = in_sizes[1] / 2;        // 6400000
  const long long* row = ei;            // edge_index[0]
  const long long* col = ei + E;        // edge_index[1]

  float* out  = (float*)d_out;          // [out (n*16) | h3 (n*16)]
  float* hfin = out + (size_t)n * 16;   // relu'd layer-3 features -> 2nd output

  // workspace layout (floats): dis[n] | h[32n] | r[32n] | agg[32n]  (~39 MB)
  float* dis = (float*)d_ws;
  float* h   = dis + n;
  float* r   = h + (size_t)32 * n;
  float* agg = r + (size_t)32 * n;

  // ---- symmetric normalization D^-1/2 ----
  k_set1 <<<cdiv(n, 256), 256, 0, stream>>>(dis, n);
  k_deg  <<<cdiv(E, 256), 256, 0, stream>>>(col, dis, E);
  k_rsqrt<<<cdiv(n, 256), 256, 0, stream>>>(dis, n);

  // ---- layer 1: 512 -> 16 ----
  k_gemm_wmma<512, 16><<<cdiv((long long)(n / 16) * 32, 128), 128, 0, stream>>>(x, W1, nullptr, h, n);
  k_zero<<<cdiv((long long)n * 16, 256), 256, 0, stream>>>(agg, n * 16);
  k_scatter<16><<<cdiv((long long)E * 16, 256), 256, 0, stream>>>(h, row, col, dis, agg, E);
  k_biasrelu<16><<<cdiv((long long)n * 16, 256), 256, 0, stream>>>(agg, h, dis, b1, r, n);

  // ---- layer 2: 16 -> 32 ----
  k_gemm_wmma<16, 32><<<cdiv((long long)(n / 16) * 2 * 32, 128), 128, 0, stream>>>(r, W2, nullptr, h, n);
  k_zero<<<cdiv((long long)n * 32, 256), 256, 0, stream>>>(agg, n * 32);
  k_scatter<32><<<cdiv((long long)E * 32, 256), 256, 0, stream>>>(h, row, col, dis, agg, E);
  k_biasrelu<32><<<cdiv((long long)n * 32, 256), 256, 0, stream>>>(agg, h, dis, b2, r, n);

  // ---- layer 3: 32 -> 16 ----
  k_gemm_wmma<32, 16><<<cdiv((long long)(n / 16) * 32, 128), 128, 0, stream>>>(r, W3, nullptr, h, n);
  k_zero<<<cdiv((long long)n * 16, 256), 256, 0, stream>>>(agg, n * 16);
  k_scatter<16><<<cdiv((long long)E * 16, 256), 256, 0, stream>>>(h, row, col, dis, agg, E);
  k_biasrelu<16><<<cdiv((long long)n * 16, 256), 256, 0, stream>>>(agg, h, dis, b3, hfin, n);

  // ---- classifier: out = h3 @ Wc + bc ----
  k_gemm_wmma<16, 16><<<cdiv((long long)(n / 16) * 32, 128), 128, 0, stream>>>(hfin, Wc, bc, out, n);
}